// MVDRBeamformer_86560770883793
// MI455X (gfx1250) — compile-verified
//
#include <hip/hip_runtime.h>
#include <hip/hip_bf16.h>

// MVDR beamformer, fully fused per covariance window.
// grid = 255 windows, block = 96 threads (3 wave32 waves, one per 16-angle tile).

typedef __attribute__((ext_vector_type(2))) float v2f;
typedef __attribute__((ext_vector_type(8))) float v8f;

#define KT 2048
#define KF 129
#define KM 8
#define KA 36
#define FC 8            // frequencies per LDS chunk
#define NCHUNK ((KF + FC - 1) / FC)

static __device__ __forceinline__ v8f wmma4(v2f a, v2f b, v8f c) {
    // D = A(16x4, f32) * B(4x16, f32) + C(16x16, f32), full precision
    return __builtin_amdgcn_wmma_f32_16x16x4_f32(
        /*neg_a=*/false, a, /*neg_b=*/false, b,
        /*c_mod=*/(short)0, c, /*reuse_a=*/false, /*reuse_b=*/false);
}

__launch_bounds__(96)
__global__ void mvdr_fused_kernel(const float* __restrict__ sre,
                                  const float* __restrict__ sim,
                                  const float* __restrict__ vre,
                                  const float* __restrict__ vim,
                                  float* __restrict__ out) {
    // LDS: snapshots (16 frames x FC freqs x 8 ant), packed Cholesky factors,
    // diag reciprocals, conj-weights padded to 48 angle rows (3 WMMA tiles).
    __shared__ __align__(16) float  lxr[16 * FC * 8];
    __shared__ __align__(16) float  lxi[16 * FC * 8];
    __shared__ __align__(16) float2 lL[FC * 36];
    __shared__ float                lDi[FC * 8];
    __shared__ __align__(16) float  lWr[FC * 48 * 8];
    __shared__ __align__(16) float  lWi[FC * 48 * 8];

    const int g    = blockIdx.x;          // window index w
    const int tid  = threadIdx.x;
    const int wave = tid >> 5;            // a-tile 0..2
    const int lane = tid & 31;
    const int col  = lane & 15;           // B col (frame) / D col / A row-within-tile
    const int h    = (lane < 16) ? 0 : 1; // K-pair selector for A/B operands

    const int t0    = (g == 0) ? 0 : (8 * g + 8);   // first output frame
    const int nt    = (g == 0) ? 16 : 8;            // frames in this window's group
    const int nsoff = (g == 0) ? 0 : 8;             // frame -> snapshot offset
    int ns = col + nsoff; if (ns > 15) ns = 15;     // clamp (cols >= nt discarded)

    const int rowA = wave * 16 + col;               // angle row 0..47 (36..47 zero)

    // zero padded weight rows 36..47 once (never rewritten)
    for (int i = tid; i < FC * 12 * 8; i += 96) {
        int fc = i / (12 * 8); int r = i % (12 * 8);
        int a = 36 + r / 8;    int m = r & 7;
        lWr[(fc * 48 + a) * 8 + m] = 0.f;
        lWi[(fc * 48 + a) * 8 + m] = 0.f;
    }

    v8f pacc = {0.f, 0.f, 0.f, 0.f, 0.f, 0.f, 0.f, 0.f};

    for (int c = 0; c < NCHUNK; ++c) {
        const int f0 = c * FC;
        const int fcmax = (KF - f0 < FC) ? (KF - f0) : FC;
        __syncthreads();  // previous chunk's readers done before refill

        // ---- phase 1: stage the 16 snapshots for this f-chunk ----
        for (int i = tid; i < 16 * fcmax * 8; i += 96) {
            int n = i / (fcmax * 8); int r = i % (fcmax * 8);
            int fc = r / 8;          int m = r & 7;
            size_t gidx = ((size_t)(8 * g + n) * KF + (f0 + fc)) * KM + m;
            lxr[(n * FC + fc) * 8 + m] = sre[gidx];
            lxi[(n * FC + fc) * 8 + m] = sim[gidx];
        }
        __syncthreads();

        // ---- phase 2: Hermitian covariance, lower triangle, task = (f, entry) ----
        for (int task = tid; task < fcmax * 36; task += 96) {
            int fc = task / 36; int e = task % 36;
            int j = 0, rem = e;
            while (rem > j) { rem -= (j + 1); ++j; }
            int k = rem;  // entry (j, k), j >= k
            float srr = 0.f, sii = 0.f;
            for (int n = 0; n < 16; ++n) {
                int base = (n * FC + fc) * 8;
                float xrj = lxr[base + j], xij = lxi[base + j];
                float xrk = lxr[base + k], xik = lxi[base + k];
                srr += xrj * xrk + xij * xik;      // Re(x_j * conj(x_k))
                sii += xij * xrk - xrj * xik;      // Im(x_j * conj(x_k))
            }
            srr *= (1.f / 16.f); sii *= (1.f / 16.f);
            if (j == k) { srr += 0.001f; sii = 0.f; }  // diagonal loading
            lL[fc * 36 + e] = make_float2(srr, sii);
        }
        __syncthreads();

        // ---- phase 3: complex Cholesky per frequency (in registers) ----
        if (tid < fcmax) {
            const int fc = tid;
            float2 Lm[36];
            #pragma unroll
            for (int e = 0; e < 36; ++e) Lm[e] = lL[fc * 36 + e];
            float dinv[8];
            #pragma unroll
            for (int j = 0; j < 8; ++j) {
                float d = Lm[j * (j + 1) / 2 + j].x;
                #pragma unroll
                for (int p = 0; p < 8; ++p) if (p < j) {
                    float2 l = Lm[j * (j + 1) / 2 + p];
                    d -= l.x * l.x + l.y * l.y;
                }
                d = fmaxf(d, 1e-20f);
                float s = sqrtf(d), inv = 1.f / s;
                Lm[j * (j + 1) / 2 + j] = make_float2(s, 0.f);
                dinv[j] = inv;
                #pragma unroll
                for (int i2 = 0; i2 < 8; ++i2) if (i2 > j) {
                    float2 cij = Lm[i2 * (i2 + 1) / 2 + j];
                    #pragma unroll
                    for (int p = 0; p < 8; ++p) if (p < j) {
                        float2 a = Lm[i2 * (i2 + 1) / 2 + p];
                        float2 b = Lm[j  * (j  + 1) / 2 + p];
                        cij.x -= a.x * b.x + a.y * b.y;   // a * conj(b)
                        cij.y -= a.y * b.x - a.x * b.y;
                    }
                    Lm[i2 * (i2 + 1) / 2 + j] = make_float2(cij.x * inv, cij.y * inv);
                }
            }
            #pragma unroll
            for (int e = 0; e < 36; ++e) lL[fc * 36 + e] = Lm[e];
            #pragma unroll
            for (int j = 0; j < 8; ++j) lDi[fc * 8 + j] = dinv[j];
        }
        __syncthreads();

        // ---- phase 4: two triangular solves + MVDR normalization, task = (f, a) ----
        for (int task = tid; task < fcmax * 36; task += 96) {
            int fc = task / 36; int a = task % 36;
            int f = f0 + fc;
            float vr[8], vi[8];
            #pragma unroll
            for (int m = 0; m < 8; ++m) {
                size_t gi = ((size_t)a * KF + f) * KM + m;
                vr[m] = vre[gi]; vi[m] = vim[gi];
            }
            float di[8];
            #pragma unroll
            for (int j = 0; j < 8; ++j) di[j] = lDi[fc * 8 + j];
            float yr2[8], yi2[8], zr[8], zi[8];
            #pragma unroll
            for (int i2 = 0; i2 < 8; ++i2) {                 // L y = v
                float ar = vr[i2], ai = vi[i2];
                #pragma unroll
                for (int k = 0; k < 8; ++k) if (k < i2) {
                    float2 l = lL[fc * 36 + i2 * (i2 + 1) / 2 + k];
                    ar -= l.x * yr2[k] - l.y * yi2[k];
                    ai -= l.x * yi2[k] + l.y * yr2[k];
                }
                yr2[i2] = ar * di[i2]; yi2[i2] = ai * di[i2];
            }
            #pragma unroll
            for (int i2 = 7; i2 >= 0; --i2) {                // L^H z = y
                float ar = yr2[i2], ai = yi2[i2];
                #pragma unroll
                for (int k = 0; k < 8; ++k) if (k > i2) {
                    float2 l = lL[fc * 36 + k * (k + 1) / 2 + i2];  // conj(L[k][i2])*z[k]
                    ar -= l.x * zr[k] + l.y * zi[k];
                    ai -= l.x * zi[k] - l.y * zr[k];
                }
                zr[i2] = ar * di[i2]; zi[i2] = ai * di[i2];
            }
            float dr = 0.f, dm = 0.f;                        // denom = v^H z
            #pragma unroll
            for (int m = 0; m < 8; ++m) {
                dr += vr[m] * zr[m] + vi[m] * zi[m];
                dm += vr[m] * zi[m] - vi[m] * zr[m];
            }
            float inv = 1.f / (dr * dr + dm * dm);
            #pragma unroll
            for (int m = 0; m < 8; ++m) {                    // store conj(z/denom)
                float wr = (zr[m] * dr + zi[m] * dm) * inv;
                float wi = (zi[m] * dr - zr[m] * dm) * inv;
                lWr[(fc * 48 + a) * 8 + m] = wr;
                lWi[(fc * 48 + a) * 8 + m] = -wi;
            }
        }
        __syncthreads();

        // ---- phase 5: WMMA beamform + power. One 16-angle tile per wave. ----
        // yr tile: [Wcr | Wci](16x16) * [xr ; -xi](16x16frames), K=16 via 4x K=4
        // yi tile: [Wci | Wcr]        * [xr ;  xi]
        for (int fc = 0; fc < fcmax; ++fc) {
            const v2f* Ar2 = (const v2f*)&lWr[(fc * 48 + rowA) * 8];
            const v2f* Ai2 = (const v2f*)&lWi[(fc * 48 + rowA) * 8];
            v2f p0 = Ar2[h],     p1 = Ar2[2 + h];            // Wcr K-pairs
            v2f q0 = Ai2[h],     q1 = Ai2[2 + h];            // Wci K-pairs
            const v2f* Br2 = (const v2f*)&lxr[(ns * FC + fc) * 8];
            const v2f* Bi2 = (const v2f*)&lxi[(ns * FC + fc) * 8];
            v2f r0 = Br2[h],     r1 = Br2[2 + h];            // xr K-pairs
            v2f s0 = Bi2[h],     s1 = Bi2[2 + h];            // xi K-pairs
            v2f n0 = -s0,        n1 = -s1;

            v8f accr = {0.f,0.f,0.f,0.f,0.f,0.f,0.f,0.f};
            v8f acci = {0.f,0.f,0.f,0.f,0.f,0.f,0.f,0.f};
            accr = wmma4(p0, r0, accr);
            accr = wmma4(p1, r1, accr);
            accr = wmma4(q0, n0, accr);
            accr = wmma4(q1, n1, accr);
            acci = wmma4(q0, r0, acci);
            acci = wmma4(q1, r1, acci);
            acci = wmma4(p0, s0, acci);
            acci = wmma4(p1, s1, acci);
            pacc += accr * accr + acci * acci;               // |y|^2, summed over f
        }
    }

    // ---- store power[a, t] from D-layout accumulators ----
    if (col < nt) {
        int t = t0 + col;
        #pragma unroll
        for (int j = 0; j < 8; ++j) {
            int a = wave * 16 + j + ((lane < 16) ? 0 : 8);
            if (a < KA) out[(size_t)a * KT + t] = pacc[j];
        }
    }
}

extern "C" void kernel_launch(void* const* d_in, const int* in_sizes, int n_in,
                              void* d_out, int out_size, void* d_ws, size_t ws_size,
                              hipStream_t stream) {
    const float* sre = (const float*)d_in[0];   // samples_re [T,F,M]
    const float* sim = (const float*)d_in[1];   // samples_im [T,F,M]
    const float* vre = (const float*)d_in[2];   // steer_re   [A,F,M]
    const float* vim = (const float*)d_in[3];   // steer_im   [A,F,M]
    float* out = (float*)d_out;                 // power      [A,T]
    (void)in_sizes; (void)n_in; (void)out_size; (void)d_ws; (void)ws_size;

    dim3 grid(255);   // one block per covariance window
    dim3 block(96);   // 3 wave32 waves, one per 16-angle WMMA tile
    mvdr_fused_kernel<<<grid, block, 0, stream>>>(sre, sim, vre, vim, out);
}